// LearnableMidiSynth_48275432407084
// MI455X (gfx1250) — compile-verified
//
#include <hip/hip_runtime.h>
#include <hip/hip_bf16.h>

// CDNA5 / gfx1250, wave32.
// Synth via rank-2 factorization onto V_WMMA_F32_16X16X4_F32:
//   sin(f*(32q+r)) = sin(32 f q) cos(f r) + cos(32 f q) sin(f r)
// Note (8192 samples) = 256x32 matrix = A(256x2) x B(2x32); each wave computes
// one 16-row slice (tm) as two 16x16 WMMA tiles (tn=0,1), then atomic-adds
// into the overlap-add output buffer (lives in L2: 16.8MB << 192MB).

typedef float v2f __attribute__((ext_vector_type(2)));
typedef float v8f __attribute__((ext_vector_type(8)));

#define NOTE_LEN    8192
#define WIN_LEN     1024
#define WIN_START   (NOTE_LEN - WIN_LEN)   // 7168
#define TM_PER_NOTE 16                     // 16 tm slices of 16 q-rows each
#define WAVES_PER_BLOCK 8                  // 256 threads

__global__ void __launch_bounds__(256) zero_f4_kernel(float4* __restrict__ out, int n4) {
    int i = blockIdx.x * blockDim.x + threadIdx.x;
    if (i < n4) {
        float4 z; z.x = 0.f; z.y = 0.f; z.z = 0.f; z.w = 0.f;
        out[i] = z;
    }
}

__global__ void __launch_bounds__(256) midi_synth_wmma_kernel(
    const float* __restrict__ freq_rad,
    const float* __restrict__ velocity,
    const float* __restrict__ window,
    const int*   __restrict__ starts,
    float*       __restrict__ out,
    int n_notes)
{
    const int lane = threadIdx.x & 31;
    // Wave id is uniform across the wave; readfirstlane makes that visible to
    // the compiler -> scalar loads for note params, s_cbranch for the window
    // path instead of per-lane exec masking.
    const int wid = __builtin_amdgcn_readfirstlane(
        blockIdx.x * WAVES_PER_BLOCK + (threadIdx.x >> 5));

    const int note = wid >> 4;         // / TM_PER_NOTE   (scalar)
    const int tm   = wid & 15;         // 16-row q slice  (scalar)
    if (note >= n_notes) return;       // uniform; grid is exact so never taken

    const float f     = freq_rad[note];                  // s_load
    const float amp   = velocity[note] * (1.0f / 127.0f);
    const int   start = starts[note];

    const int  l16 = lane & 15;
    const bool lo  = (lane < 16);

    // ---- A operand: lanes 0-15: K0 = amp*sin(32 f q), K1 = amp*cos(32 f q);
    //      lanes 16-31 (K2,K3) = 0
    const int   q0  = tm * 16 + l16;
    const float phA = f * (float)(q0 << 5);      // 32*f*q
    const float sA  = __sinf(phA);
    const float cA  = __cosf(phA);
    v2f a;
    a.x = lo ? amp * sA : 0.0f;
    a.y = lo ? amp * cA : 0.0f;

    // ---- B operands for tn=0 (r=0..15) and tn=1 (r=16..31):
    //      row K0 = cos(f r), row K1 = sin(f r); rows K2,K3 = 0 (lanes 16-31)
    const float phB0 = f * (float)l16;
    const float phB1 = f * (float)(l16 + 16);
    v2f b0, b1;
    b0.x = lo ? __cosf(phB0) : 0.0f;
    b0.y = lo ? __sinf(phB0) : 0.0f;
    b1.x = lo ? __cosf(phB1) : 0.0f;
    b1.y = lo ? __sinf(phB1) : 0.0f;

    v8f c = {};
    // D[m][n] = amp * sin(f * (32*(16*tm + m) + 16*tn + n))
    v8f d0 = __builtin_amdgcn_wmma_f32_16x16x4_f32(false, a, false, b0, (short)0, c, false, false);
    v8f d1 = __builtin_amdgcn_wmma_f32_16x16x4_f32(false, a, false, b1, (short)0, c, false, false);

    // ---- Scatter-add; Hann window applies only to n >= 7168 <=> tm >= 14,
    // which is a *scalar* condition -> single s_cbranch, two straight loops.
    // D layout: lane L holds N = L&15; element j is row M = j + 8*(L>=16).
    const int  mAdd   = lo ? 0 : 8;
    const int  N      = l16;
    float* const outp = out + start;

    if (tm >= 14) {
        #pragma unroll
        for (int j = 0; j < 8; ++j) {
            const int q  = tm * 16 + j + mAdd;
            const int n0 = (q << 5) + N;       // tn = 0
            const int n1 = n0 + 16;            // tn = 1
            const float v0 = d0[j] * window[n0 - WIN_START];
            const float v1 = d1[j] * window[n1 - WIN_START];
            __hip_atomic_fetch_add(outp + n0, v0, __ATOMIC_RELAXED, __HIP_MEMORY_SCOPE_AGENT);
            __hip_atomic_fetch_add(outp + n1, v1, __ATOMIC_RELAXED, __HIP_MEMORY_SCOPE_AGENT);
        }
    } else {
        #pragma unroll
        for (int j = 0; j < 8; ++j) {
            const int q  = tm * 16 + j + mAdd;
            const int n0 = (q << 5) + N;       // tn = 0
            const int n1 = n0 + 16;            // tn = 1
            __hip_atomic_fetch_add(outp + n0, d0[j], __ATOMIC_RELAXED, __HIP_MEMORY_SCOPE_AGENT);
            __hip_atomic_fetch_add(outp + n1, d1[j], __ATOMIC_RELAXED, __HIP_MEMORY_SCOPE_AGENT);
        }
    }
}

extern "C" void kernel_launch(void* const* d_in, const int* in_sizes, int n_in,
                              void* d_out, int out_size, void* d_ws, size_t ws_size,
                              hipStream_t stream) {
    const float* freq_rad = (const float*)d_in[0];
    const float* velocity = (const float*)d_in[1];
    const float* window   = (const float*)d_in[2];
    const int*   starts   = (const int*)d_in[3];
    // d_in[4] = note_len (8192), d_in[5] = duration_samples — fixed by the
    // reference; the 256x32 WMMA tiling hardcodes note_len=8192, WIN_LEN=1024.

    float* out = (float*)d_out;
    const int n_notes = in_sizes[0];

    // 1) zero the accumulation buffer (harness poisons d_out; we accumulate)
    const int n4 = out_size >> 2;
    zero_f4_kernel<<<(n4 + 255) / 256, 256, 0, stream>>>((float4*)out, n4);

    // 2) WMMA synth + overlap-add: one wave per (note, tm) slice
    const int total_waves = n_notes * TM_PER_NOTE;              // 65536
    const int blocks = (total_waves + WAVES_PER_BLOCK - 1) / WAVES_PER_BLOCK; // 8192
    midi_synth_wmma_kernel<<<blocks, 256, 0, stream>>>(
        freq_rad, velocity, window, starts, out, n_notes);
}